// MultiBoxLoss_64922725646455
// MI455X (gfx1250) — compile-verified
//
#include <hip/hip_runtime.h>
#include <float.h>

// MultiBox (SSD) loss for MI455X / gfx1250 (wave32).
// Memory-bound: dominant cost is one streaming pass over conf_preds (~509MB -> ~22us @ 23.3TB/s).
// Determinism: no float atomics; integer/u64-max atomics only; float sums go through
// per-block partials reduced in fixed order by a V_WMMA_F32_16X16X4_F32 ones-matmul.

typedef __attribute__((ext_vector_type(2))) float v2f;
typedef __attribute__((ext_vector_type(8))) float v8f;

#define IOU_THR   0.5f
#define NEG_RATIO 3

__device__ __forceinline__ float warp_max_f(float v) {
#pragma unroll
  for (int m = 16; m; m >>= 1) v = fmaxf(v, __shfl_xor(v, m, 32));
  return v;
}
__device__ __forceinline__ float warp_sum_f(float v) {
#pragma unroll
  for (int m = 16; m; m >>= 1) v += __shfl_xor(v, m, 32);
  return v;
}

// ---------------------------------------------------------------- zero helper
__global__ void k_zero32(unsigned* __restrict__ p, int n) {
  int i = blockIdx.x * 256 + threadIdx.x;
  if (i < n) p[i] = 0u;
}

// ---------------------------------------------------------------- K1: IoU match
// One block = 256 priors of one batch row. GT boxes cached in LDS.
// Per-(b,m) best prior tracked with packed u64 (iou_bits<<32 | ~p): max => best iou,
// ties => smallest prior index (matches jnp.argmax first-occurrence). Deterministic.
__global__ void k_iou(const float* __restrict__ priors, const float* __restrict__ gtb,
                      float* __restrict__ bestOv, int* __restrict__ bestIdx,
                      unsigned long long* __restrict__ bestPrior, int P, int M) {
  int b = blockIdx.y;
  int p = blockIdx.x * 256 + threadIdx.x;
  __shared__ float g[128];                 // up to 32 boxes * 4
  __shared__ unsigned long long key[32];
  if (threadIdx.x < M) key[threadIdx.x] = 0ull;
  if (threadIdx.x < M * 4) g[threadIdx.x] = gtb[(size_t)b * M * 4 + threadIdx.x];
  __syncthreads();
  if (p < P) {
    float cx = priors[p * 4 + 0], cy = priors[p * 4 + 1];
    float w = priors[p * 4 + 2], h = priors[p * 4 + 3];
    float ax1 = cx - 0.5f * w, ay1 = cy - 0.5f * h;
    float ax2 = cx + 0.5f * w, ay2 = cy + 0.5f * h;
    float areaA = (ax2 - ax1) * (ay2 - ay1);
    float best = -1.0f; int bi = 0;
    for (int m = 0; m < M; m++) {
      float bx1 = g[m * 4 + 0], by1 = g[m * 4 + 1], bx2 = g[m * 4 + 2], by2 = g[m * 4 + 3];
      float ix = fmaxf(fminf(ax2, bx2) - fmaxf(ax1, bx1), 0.0f);
      float iy = fmaxf(fminf(ay2, by2) - fmaxf(ay1, by1), 0.0f);
      float inter = ix * iy;
      float iou = inter / (areaA + (bx2 - bx1) * (by2 - by1) - inter);
      if (iou > best) { best = iou; bi = m; }
      unsigned long long kk =
          ((unsigned long long)__float_as_uint(iou) << 32) | (unsigned)(~(unsigned)p);
      if (kk > key[m]) atomicMax(&key[m], kk);   // DS_MAX_U64, order-independent
    }
    bestOv[(size_t)b * P + p] = best;
    bestIdx[(size_t)b * P + p] = bi;
  }
  __syncthreads();
  if (threadIdx.x < M && key[threadIdx.x] != 0ull)
    atomicMax(&bestPrior[b * M + threadIdx.x], key[threadIdx.x]);
}

// ---------------------------------------------------------------- K2: force-match
// Serial over m per row => last-write-wins on duplicate priors (scatter .set semantics).
__global__ void k_force(const unsigned long long* __restrict__ bestPrior,
                        float* __restrict__ bestOv, int* __restrict__ bestIdx,
                        int P, int M, int B) {
  int b = blockIdx.x * blockDim.x + threadIdx.x;
  if (b >= B) return;
  for (int m = 0; m < M; m++) {
    unsigned long long kk = bestPrior[b * M + m];
    unsigned p = ~(unsigned)(kk & 0xFFFFFFFFu);
    bestIdx[(size_t)b * P + p] = m;
    bestOv[(size_t)b * P + p] = 1.0f;
  }
}

// ---------------------------------------------------------------- K3: loc loss + pos counts
__global__ void k_loc(const float* __restrict__ locp, const float* __restrict__ priors,
                      const float* __restrict__ gtb, const float* __restrict__ bestOv,
                      const int* __restrict__ bestIdx, float* __restrict__ locPart,
                      int* __restrict__ rowPos, int* __restrict__ posTotal, int P, int M) {
  int b = blockIdx.y;
  int p = blockIdx.x * 256 + threadIdx.x;
  float acc = 0.0f; int pc = 0;
  if (p < P) {
    size_t li = (size_t)b * P + p;
    float ov = bestOv[li];
    if (ov > IOU_THR) {
      pc = 1;
      int m = bestIdx[li];
      const float* g = &gtb[((size_t)b * M + m) * 4];
      float gx1 = g[0], gy1 = g[1], gx2 = g[2], gy2 = g[3];
      float pcx = priors[p * 4 + 0], pcy = priors[p * 4 + 1];
      float pw = priors[p * 4 + 2], ph = priors[p * 4 + 3];
      float e0 = (0.5f * (gx1 + gx2) - pcx) / pw;
      float e1 = (0.5f * (gy1 + gy2) - pcy) / ph;
      float e2 = __logf((gx2 - gx1) / pw);
      float e3 = __logf((gy2 - gy1) / ph);
      const float* lp = &locp[li * 4];
      float d0 = lp[0] - e0, d1 = lp[1] - e1, d2 = lp[2] - e2, d3 = lp[3] - e3;
#define SL1(d) (fabsf(d) < 1.0f ? 0.5f * (d) * (d) : fabsf(d) - 0.5f)
      acc = SL1(d0) + SL1(d1) + SL1(d2) + SL1(d3);
#undef SL1
    }
  }
  __shared__ float sf[256];
  __shared__ int si[256];
  sf[threadIdx.x] = acc; si[threadIdx.x] = pc;
  __syncthreads();
  for (int s = 128; s; s >>= 1) {
    if (threadIdx.x < s) { sf[threadIdx.x] += sf[threadIdx.x + s]; si[threadIdx.x] += si[threadIdx.x + s]; }
    __syncthreads();
  }
  if (threadIdx.x == 0) {
    locPart[blockIdx.y * gridDim.x + blockIdx.x] = sf[0];
    if (si[0]) { atomicAdd(&rowPos[b], si[0]); atomicAdd(posTotal, si[0]); }
  }
}

// ---------------------------------------------------------------- K4: cross-entropy stream
// Wave-per-prior: 32 lanes read 81 contiguous f32 (coalesced ~324B/wave), warp-reduce
// max + sum-exp. Prefetch keeps the 509MB stream ahead.
__global__ void k_ce(const float* __restrict__ conf, const float* __restrict__ bestOv,
                     const int* __restrict__ bestIdx, const int* __restrict__ labels,
                     float* __restrict__ ceNeg, float* __restrict__ cePosPart,
                     int P, int M, int C, int B, int nblocks) {
  int wid = threadIdx.x >> 5, lane = threadIdx.x & 31;
  int wavesPer = blockDim.x >> 5;
  long long total = (long long)B * P;
  long long stride = (long long)nblocks * wavesPer;
  float acc = 0.0f;
  for (long long base = (long long)blockIdx.x * wavesPer + wid; base < total; base += stride) {
    const float* row = &conf[base * C];
    __builtin_prefetch(row + stride * C, 0, 1);     // global_prefetch_b8
    float x0 = (lane < C)      ? row[lane]      : -FLT_MAX;
    float x1 = (lane + 32 < C) ? row[lane + 32] : -FLT_MAX;
    float x2 = (lane + 64 < C) ? row[lane + 64] : -FLT_MAX;
    float mx = warp_max_f(fmaxf(x0, fmaxf(x1, x2)));
    float s = ((lane < C)      ? __expf(x0 - mx) : 0.0f) +
              ((lane + 32 < C) ? __expf(x1 - mx) : 0.0f) +
              ((lane + 64 < C) ? __expf(x2 - mx) : 0.0f);
    s = warp_sum_f(s);
    int b = (int)(base / P);
    float ov = bestOv[base];
    bool pos = ov > IOU_THR;
    int t = pos ? labels[b * M + bestIdx[base]] : 0;
    float ce = __logf(s) + mx - row[t];
    if (lane == 0) {
      ceNeg[base] = pos ? 0.0f : fmaxf(ce, 0.0f);   // clamp keeps u32-bit order monotonic
      if (pos) acc += ce;
    }
  }
  __shared__ float sf[256];
  sf[threadIdx.x] = acc;
  __syncthreads();
  for (int s2 = 128; s2; s2 >>= 1) {
    if (threadIdx.x < s2) sf[threadIdx.x] += sf[threadIdx.x + s2];
    __syncthreads();
  }
  if (threadIdx.x == 0) cePosPart[blockIdx.x] = sf[0];
}

// ---------------------------------------------------------------- radix select (exact top-k)
__global__ void k_scan_init(const int* __restrict__ rowPos, int* __restrict__ kRem,
                            unsigned* __restrict__ prefix, int B, int P) {
  int b = blockIdx.x * blockDim.x + threadIdx.x;
  if (b >= B) return;
  int k = rowPos[b] * NEG_RATIO;
  if (k < 1) k = 1;
  if (k > P) k = P;
  kRem[b] = k; prefix[b] = 0u;
}

__global__ void k_hist(const float* __restrict__ ceNeg, const unsigned* __restrict__ prefix,
                       unsigned* __restrict__ hist, int P, unsigned maskHi, int shift) {
  int b = blockIdx.y;
  int p = blockIdx.x * 256 + threadIdx.x;
  __shared__ unsigned h[256];
  h[threadIdx.x] = 0u;
  __syncthreads();
  unsigned pref = prefix[b];
  if (p < P) {
    unsigned bits = __float_as_uint(ceNeg[(size_t)b * P + p]);
    if ((bits & maskHi) == pref) atomicAdd(&h[(bits >> shift) & 0xFF], 1u);
  }
  __syncthreads();
  if (h[threadIdx.x]) atomicAdd(&hist[b * 256 + threadIdx.x], h[threadIdx.x]);
}

__global__ void k_scan(unsigned* __restrict__ hist, unsigned* __restrict__ prefix,
                       int* __restrict__ kRem, int B, int shift) {
  int b = blockIdx.x * blockDim.x + threadIdx.x;
  if (b >= B) return;
  int k = kRem[b];
  unsigned cum = 0; int chosen = 0;
  for (int bin = 255; bin >= 0; bin--) {
    unsigned c = hist[b * 256 + bin];
    if (cum + c >= (unsigned)k) { chosen = bin; break; }
    cum += c;
  }
  prefix[b] |= (unsigned)chosen << shift;
  kRem[b] = k - (int)cum;
  for (int bin = 0; bin < 256; bin++) hist[b * 256 + bin] = 0u;  // ready for next pass
}

__global__ void k_negsum(const float* __restrict__ ceNeg, const unsigned* __restrict__ prefix,
                         float* __restrict__ negPart, int P) {
  int b = blockIdx.y;
  int p = blockIdx.x * 256 + threadIdx.x;
  unsigned T = prefix[b];
  float v = 0.0f;
  if (p < P) {
    float x = ceNeg[(size_t)b * P + p];
    if (__float_as_uint(x) > T) v = x;   // strictly greater; ties handled as kRem*T
  }
  __shared__ float sf[256];
  sf[threadIdx.x] = v;
  __syncthreads();
  for (int s = 128; s; s >>= 1) {
    if (threadIdx.x < s) sf[threadIdx.x] += sf[threadIdx.x + s];
    __syncthreads();
  }
  if (threadIdx.x == 0) negPart[blockIdx.y * gridDim.x + blockIdx.x] = sf[0];
}

// ---------------------------------------------------------------- finalize (WMMA f32 reduce)
// Single wave32, EXEC all-ones. A = 16x4 f32 data tile, B = ones(4x16) => D rows are exact
// f32 row-sums accumulated through C in fixed order (deterministic).
// Main loop is unguarded (full 64-float chunks, one b64 load/lane/iter); only the final
// partial chunk (if any) takes the guarded path.
__device__ float wmma_sum(const float* __restrict__ arr, int n, int lane) {
  v8f c = {};
  v2f bones; bones.x = 1.0f; bones.y = 1.0f;
  int m = lane & 15;       // A row
  int khalf = lane >> 4;   // K pair {0,1} or {2,3}
  int laneOff = 4 * m + 2 * khalf;       // A[m][k] = arr[base + 4m + k]
  int nfull = n & ~63;
  for (int base = 0; base < nfull; base += 64) {
    v2f a = *(const v2f*)(arr + base + laneOff);   // global_load_b64, no exec games
    c = __builtin_amdgcn_wmma_f32_16x16x4_f32(false, a, false, bones, (short)0, c,
                                              false, false);
  }
  if (nfull < n) {                        // tail chunk (absent when n % 64 == 0)
    int i0 = nfull + laneOff;
    v2f a;
    a.x = (i0     < n) ? arr[i0]     : 0.0f;
    a.y = (i0 + 1 < n) ? arr[i0 + 1] : 0.0f;
    c = __builtin_amdgcn_wmma_f32_16x16x4_f32(false, a, false, bones, (short)0, c,
                                              false, false);
  }
  float s = c[0] + c[1] + c[2] + c[3] + c[4] + c[5] + c[6] + c[7];
  // column 0: rows 0..7 live in lane 0, rows 8..15 in lane 16
  return __shfl(s, 0, 32) + __shfl(s, 16, 32);
}

__global__ void k_final(const float* __restrict__ locPart, int nLoc,
                        const float* __restrict__ cePart, int nCe,
                        const int* __restrict__ kRem, const unsigned* __restrict__ prefix,
                        const int* __restrict__ posTotal, int B, float* __restrict__ out) {
  int lane = threadIdx.x;
  float locSum = wmma_sum(locPart, nLoc, lane);
  float ceSum = wmma_sum(cePart, nCe, lane);
  float tie = 0.0f;
  for (int b = lane; b < B; b += 32)
    tie += (float)kRem[b] * __uint_as_float(prefix[b]);
  tie = warp_sum_f(tie);
  if (lane == 0) {
    int np = *posTotal; if (np < 1) np = 1;
    float fnp = (float)np;
    out[0] = locSum / fnp + (ceSum + tie) / fnp;
  }
}

// ---------------------------------------------------------------- host launcher
extern "C" void kernel_launch(void* const* d_in, const int* in_sizes, int n_in,
                              void* d_out, int out_size, void* d_ws, size_t ws_size,
                              hipStream_t stream) {
  const float* locp   = (const float*)d_in[0];
  const float* conf   = (const float*)d_in[1];
  const float* priors = (const float*)d_in[2];
  const float* gtb    = (const float*)d_in[3];
  const int*   labels = (const int*)d_in[4];

  int P = in_sizes[2] / 4;
  int B = (int)((long long)in_sizes[0] / ((long long)P * 4));
  int C = (int)((long long)in_sizes[1] / ((long long)B * P));
  int M = in_sizes[4] / B;
  long long BP = (long long)B * P;
  int tiles = (P + 255) / 256;
  const int NB4 = 4096;

  // bump-allocate workspace (~19 MB total)
  char* ws = (char*)d_ws; size_t off = 0;
  auto alloc = [&](size_t bytes) -> void* {
    void* q = ws + off;
    off = (off + bytes + 255) & ~(size_t)255;
    return q;
  };
  int*      bestIdx   = (int*)alloc(BP * 4);
  float*    bestOv    = (float*)alloc(BP * 4);
  float*    ceNeg     = (float*)alloc(BP * 4);
  unsigned long long* bestPrior = (unsigned long long*)alloc((size_t)B * M * 8);
  int*      rowPos    = (int*)alloc((size_t)B * 4);
  unsigned* prefix    = (unsigned*)alloc((size_t)B * 4);
  int*      kRem      = (int*)alloc((size_t)B * 4);
  int*      posTotal  = (int*)alloc(4);
  unsigned* hist      = (unsigned*)alloc((size_t)B * 256 * 4);
  float*    locPart   = (float*)alloc((size_t)B * tiles * 4);
  float*    cePart    = (float*)alloc((size_t)(NB4 + B * tiles) * 4);  // [cePos | negSum] contiguous
  (void)ws_size; (void)n_in; (void)out_size;

  // init (every read buffer is re-initialized each call; d_ws may be poisoned)
  int nBP2 = B * M * 2;
  k_zero32<<<(nBP2 + 255) / 256, 256, 0, stream>>>((unsigned*)bestPrior, nBP2);
  k_zero32<<<(B + 255) / 256, 256, 0, stream>>>((unsigned*)rowPos, B);
  k_zero32<<<1, 256, 0, stream>>>((unsigned*)posTotal, 1);
  k_zero32<<<(B * 256 + 255) / 256, 256, 0, stream>>>(hist, B * 256);

  dim3 gridBP(tiles, B);
  k_iou<<<gridBP, 256, 0, stream>>>(priors, gtb, bestOv, bestIdx, bestPrior, P, M);
  k_force<<<(B + 63) / 64, 64, 0, stream>>>(bestPrior, bestOv, bestIdx, P, M, B);
  k_loc<<<gridBP, 256, 0, stream>>>(locp, priors, gtb, bestOv, bestIdx, locPart,
                                    rowPos, posTotal, P, M);
  k_ce<<<NB4, 256, 0, stream>>>(conf, bestOv, bestIdx, labels, ceNeg, cePart,
                                P, M, C, B, NB4);
  k_scan_init<<<(B + 63) / 64, 64, 0, stream>>>(rowPos, kRem, prefix, B, P);

  for (int pass = 0; pass < 4; pass++) {
    int shift = 24 - 8 * pass;
    unsigned maskHi = (pass == 0) ? 0u : (0xFFFFFFFFu << (shift + 8));
    k_hist<<<gridBP, 256, 0, stream>>>(ceNeg, prefix, hist, P, maskHi, shift);
    k_scan<<<(B + 63) / 64, 64, 0, stream>>>(hist, prefix, kRem, B, shift);
  }

  k_negsum<<<gridBP, 256, 0, stream>>>(ceNeg, prefix, cePart + NB4, P);
  k_final<<<1, 32, 0, stream>>>(locPart, B * tiles, cePart, NB4 + B * tiles,
                                kRem, prefix, posTotal, B, (float*)d_out);
}